// GatedBlock_20151986553159
// MI455X (gfx1250) — compile-verified
//
#include <hip/hip_runtime.h>

typedef __attribute__((ext_vector_type(2))) float v2f;
typedef __attribute__((ext_vector_type(8))) float v8f;

#define BATCH 32
#define DIM   2048
#define NBLK  16
#define KDROP 8   // NB/2 gates dropped per sample

// ---------------------------------------------------------------------------
// Kernel 1: gate network.  g = sigmoid(x @ gate_w + gate_b), zero the 8
// smallest gates per sample (stable rank => matches jax.lax.top_k(-g) ties).
// 512 threads: thread t -> (sample b = t>>4, gate j = t&15).
// ---------------------------------------------------------------------------
__global__ __launch_bounds__(512)
void gate_kernel(const float* __restrict__ x,
                 const float* __restrict__ gate_w,
                 const float* __restrict__ gate_b,
                 float* __restrict__ gates /* [BATCH][NBLK] */) {
  __shared__ float sg[BATCH][NBLK];
  const int t = threadIdx.x;
  const int b = t >> 4;
  const int j = t & 15;

  float acc = 0.0f;
  const float* xr = x + b * DIM;
#pragma unroll 4
  for (int d = 0; d < DIM; ++d) {
    acc += xr[d] * gate_w[d * NBLK + j];   // gate_w is (D, NB) row-major
  }
  const float g = 1.0f / (1.0f + __expf(-(acc + gate_b[j])));
  sg[b][j] = g;
  __syncthreads();

  // stable rank among the 16 gates of this sample; rank < KDROP => dropped
  int rank = 0;
#pragma unroll
  for (int k = 0; k < NBLK; ++k) {
    const float v = sg[b][k];
    if (v < g || (v == g && k < j)) ++rank;
  }
  gates[b * NBLK + j] = (rank < KDROP) ? 0.0f : g;
}

// ---------------------------------------------------------------------------
// Kernel 2: out[b,i] = gates[b, i>>7] * (weight[i,:] . x[b,:]) + bias[i]
// via V_WMMA_F32_16X16X4_F32.  M=batch tile (16), N=output-dim tile (16),
// K=input dim (2048, stepped by 4 per WMMA, two interleaved accumulators).
//
// fp32 fragment layouts (ISA 7.12.2):
//   A 16x4 : lane<16 holds A[lane][0..1] in v0/v1; lane>=16 holds A[lane-16][2..3]
//   B 4x16 : lane<16 holds B[0..1][lane];          lane>=16 holds B[2..3][lane-16]
//   C/D    : vgpr v, lane<16 -> D[v][lane]; lane>=16 -> D[v+8][lane-16]
// Per lane both A and B fragments are two consecutive floats -> float2 loads.
// ---------------------------------------------------------------------------
__global__ __launch_bounds__(128)
void gated_gemm_wmma(const float* __restrict__ x,      // [BATCH][DIM]
                     const float* __restrict__ w,      // [DIM][DIM] (out, in)
                     const float* __restrict__ bias,   // [DIM]
                     const float* __restrict__ gates,  // [BATCH][NBLK]
                     float* __restrict__ out) {        // [BATCH][DIM]
  const int lane = threadIdx.x & 31;
  const int wv   = threadIdx.x >> 5;                // wave in block (0..3)
  const int gwid = blockIdx.x * 4 + wv;             // 0..255
  const int mt   = gwid & 1;                        // batch tile (0..1)
  const int nt   = gwid >> 1;                       // output tile (0..127)
  const int n0   = nt * 16;
  const int h    = lane >> 4;                       // lane half
  const int lr   = lane & 15;

  // A: rows = batches mt*16+lr ; cols k + 2h, k + 2h + 1
  const float* xr = x + (mt * 16 + lr) * DIM + h * 2;
  // B: rows (of weight) = output dims n0+lr ; same column pairing
  const float* wr = w + (n0 + lr) * DIM + h * 2;

  v8f acc_e = {};
  v8f acc_o = {};
#pragma unroll 2
  for (int k = 0; k < DIM; k += 8) {
    const v2f ae = *(const v2f*)(xr + k);
    const v2f be = *(const v2f*)(wr + k);
    const v2f ao = *(const v2f*)(xr + k + 4);
    const v2f bo = *(const v2f*)(wr + k + 4);
    // 8 args: (neg_a, A, neg_b, B, c_mod, C, reuse_a, reuse_b)
    acc_e = __builtin_amdgcn_wmma_f32_16x16x4_f32(
        false, ae, false, be, (short)0, acc_e, false, false);
    acc_o = __builtin_amdgcn_wmma_f32_16x16x4_f32(
        false, ao, false, bo, (short)0, acc_o, false, false);
  }

  const int   col = n0 + lr;      // constant i>>7 within a 16-wide tile
  const int   blk = n0 >> 7;
  const float bv  = bias[col];
#pragma unroll
  for (int v = 0; v < 8; ++v) {
    const int   row = mt * 16 + v + 8 * h;          // batch index
    const float g   = gates[row * NBLK + blk];
    out[row * DIM + col] = g * (acc_e[v] + acc_o[v]) + bv;
  }
}

// ---------------------------------------------------------------------------
extern "C" void kernel_launch(void* const* d_in, const int* in_sizes, int n_in,
                              void* d_out, int out_size, void* d_ws, size_t ws_size,
                              hipStream_t stream) {
  const float* x      = (const float*)d_in[0];  // (32, 2048)
  const float* gate_w = (const float*)d_in[1];  // (2048, 16)
  const float* gate_b = (const float*)d_in[2];  // (16,)
  const float* weight = (const float*)d_in[3];  // (2048, 2048)
  const float* bias   = (const float*)d_in[4];  // (2048,)
  float*       out    = (float*)d_out;          // (32, 2048)
  float*       gates  = (float*)d_ws;           // 32*16 floats scratch

  gate_kernel<<<1, 512, 0, stream>>>(x, gate_w, gate_b, gates);
  gated_gemm_wmma<<<64, 128, 0, stream>>>(x, weight, bias, gates, out);
}